// dummy_gnn_model_87686052315764
// MI455X (gfx1250) — compile-verified
//
#include <hip/hip_runtime.h>

// ---------------------------------------------------------------------------
// GNN (3-layer GraphSAGE-like) for MI455X / gfx1250.
// Memory-bound: gather/scatter-mean over 1.6M edges dominates; dense layers
// use fp32 WMMA (v_wmma_f32_16x16x4_f32) to keep the GEMM on the matrix pipe
// at full fp32 precision (GEMM is ~8 GFLOP total -> never the bottleneck).
// ---------------------------------------------------------------------------

typedef __attribute__((ext_vector_type(2))) float v2f;
typedef __attribute__((ext_vector_type(8))) float v8f;

#define N_NODES 50000
#define N_EDGES 1600000
#define DIM     128          // IN_DIM == HID_DIM == 128
#define KDIM    256          // concat(agg, h) width

// ---------------------------------------------------------------------------
// Degree accumulation: one thread per edge.
// ---------------------------------------------------------------------------
__global__ void deg_kernel(const int* __restrict__ dst, float* __restrict__ deg,
                           int E) {
    int e = blockIdx.x * blockDim.x + threadIdx.x;
    if (e < E) atomicAdd(&deg[dst[e]], 1.0f);
}

__global__ void invdeg_kernel(const float* __restrict__ deg,
                              float* __restrict__ inv, int N) {
    int i = blockIdx.x * blockDim.x + threadIdx.x;
    if (i < N) {
        float d = deg[i];
        inv[i] = (d > 0.0f) ? (1.0f / d) : 0.0f;
    }
}

// ---------------------------------------------------------------------------
// Edge scatter: agg[dst] += h[src] * w.  One wave32 per edge; each lane moves
// 4 contiguous floats (float4 load = global_load_b128; 32 lanes cover the
// full 512B row), then 4 global f32 atomics.
// ---------------------------------------------------------------------------
__global__ __launch_bounds__(256)
void scatter_kernel(const float* __restrict__ h,
                    const float* __restrict__ ew,
                    const int* __restrict__ src,
                    const int* __restrict__ dst,
                    float* __restrict__ agg, int E) {
    int tid  = blockIdx.x * blockDim.x + threadIdx.x;
    int e    = tid >> 5;
    int lane = tid & 31;
    if (e >= E) return;

    int   s = src[e];
    int   d = dst[e];
    float w = ew[e];

    const float4* hv = (const float4*)(h + (size_t)s * DIM);
    float4 v = hv[lane];

    float* out = agg + (size_t)d * DIM + lane * 4;
    atomicAdd(out + 0, v.x * w);
    atomicAdd(out + 1, v.y * w);
    atomicAdd(out + 2, v.z * w);
    atomicAdd(out + 3, v.w * w);
}

// ---------------------------------------------------------------------------
// Dense layer: out[i, j] = relu( sum_k X[i,k] * W[j,k] + b[j] )
//   X[i, 0:128)   = agg[i,:] * inv_deg[i]
//   X[i, 128:256) = hprev[i,:]
// One block handles 16 nodes; wave w handles output columns [16w, 16w+16).
// K-loop of 64 x v_wmma_f32_16x16x4_f32 per wave.
//
// Fragment layouts (ISA 7.12.2, 32-bit, wave32):
//   A 16x4 : lane<16 -> M=lane, a[0]=K0,a[1]=K1 ; lane>=16 -> a[0]=K2,a[1]=K3
//   B 4x16 : lane<16 -> N=lane, b[0]=K0,b[1]=K1 ; lane>=16 -> b[0]=K2,b[1]=K3
//   C/D    : vgpr r, lane<16 -> (M=r,   N=lane)
//                    lane>=16-> (M=r+8, N=lane-16)
// ---------------------------------------------------------------------------
template <int D_OUT>
__global__ __launch_bounds__(32 * (D_OUT / 16))
void gemm_concat_kernel(const float* __restrict__ agg,
                        const float* __restrict__ hprev,
                        const float* __restrict__ inv_deg,
                        const float* __restrict__ W,     // [D_OUT, 256]
                        const float* __restrict__ bias,  // [D_OUT]
                        float* __restrict__ out) {       // [N, D_OUT]
    constexpr int NWAVES  = D_OUT / 16;
    constexpr int NTHREAD = 32 * NWAVES;
    constexpr int LDSROW  = KDIM + 2;  // pad to stagger banks

    __shared__ float xs[16][LDSROW];

    const int row0 = blockIdx.x * 16;
    const int t    = threadIdx.x;

    // Stage the 16x256 concat tile into LDS, coalesced along k.
    for (int idx = t; idx < 16 * KDIM; idx += NTHREAD) {
        int r = idx >> 8;      // / 256
        int c = idx & 255;     // % 256
        int node = row0 + r;
        float v;
        if (c < DIM) v = agg[(size_t)node * DIM + c] * inv_deg[node];
        else         v = hprev[(size_t)node * DIM + (c - DIM)];
        xs[r][c] = v;
    }
    __syncthreads();

    const int wave = t >> 5;
    const int lane = t & 31;
    const int half = lane >> 4;   // 0: K pair {0,1}, 1: K pair {2,3}
    const int m    = lane & 15;   // A row / B col within tile
    const int n0   = wave * 16;   // output column tile

    const float* wrow = W + (size_t)(n0 + m) * KDIM;  // B: lane indexes N

    v8f acc = {};
#pragma unroll 8
    for (int k = 0; k < KDIM; k += 4) {
        const int kc = k + 2 * half;
        v2f a;
        a.x = xs[m][kc];
        a.y = xs[m][kc + 1];
        v2f b;
        b.x = wrow[kc];
        b.y = wrow[kc + 1];
        acc = __builtin_amdgcn_wmma_f32_16x16x4_f32(
            /*neg_a=*/false, a, /*neg_b=*/false, b,
            /*c_mod=*/(short)0, acc, /*reuse_a=*/false, /*reuse_b=*/false);
    }

    // Epilogue: bias + ReLU, C/D layout -> row-major store.
    const float bb = bias[n0 + m];
#pragma unroll
    for (int r = 0; r < 8; ++r) {
        int row = row0 + r + 8 * half;
        float v = acc[r] + bb;
        out[(size_t)row * D_OUT + (n0 + m)] = fmaxf(v, 0.0f);
    }
}

// ---------------------------------------------------------------------------
// Host-side orchestration (graph-capture safe: only async ops on `stream`).
// ---------------------------------------------------------------------------
extern "C" void kernel_launch(void* const* d_in, const int* in_sizes, int n_in,
                              void* d_out, int out_size, void* d_ws, size_t ws_size,
                              hipStream_t stream) {
    const float* n_feat = (const float*)d_in[0];   // [N, 128]
    const float* ew     = (const float*)d_in[1];   // [E]
    const int*   src    = (const int*)  d_in[2];   // [E]
    const int*   dst    = (const int*)  d_in[3];   // [E]
    const float* W1     = (const float*)d_in[4];   // [128, 256]
    const float* b1     = (const float*)d_in[5];
    const float* W2     = (const float*)d_in[6];   // [128, 256]
    const float* b2     = (const float*)d_in[7];
    const float* W3     = (const float*)d_in[8];   // [64, 256]
    const float* b3     = (const float*)d_in[9];

    const int N = N_NODES;
    const int E = N_EDGES;

    // Workspace carve-out.
    float* ws   = (float*)d_ws;
    float* agg  = ws;                         // N*128
    float* h1   = agg + (size_t)N * DIM;      // N*128
    float* h2   = h1  + (size_t)N * DIM;      // N*128
    float* deg  = h2  + (size_t)N * DIM;      // N
    float* inv  = deg + N;                    // N

    const int edgeThreads  = 256;
    const int edgeBlocks1  = (E + edgeThreads - 1) / edgeThreads;          // 1 thread/edge
    const int edgeBlocks32 = (E * 32 + edgeThreads - 1) / edgeThreads;     // 1 wave/edge
    const int nodeBlocks   = (N + edgeThreads - 1) / edgeThreads;
    const int tileBlocks   = N / 16;  // 50000/16 = 3125 exactly

    // Degrees (shared by all layers).
    hipMemsetAsync(deg, 0, (size_t)N * sizeof(float), stream);
    deg_kernel<<<edgeBlocks1, edgeThreads, 0, stream>>>(dst, deg, E);
    invdeg_kernel<<<nodeBlocks, edgeThreads, 0, stream>>>(deg, inv, N);

    // Layer 1: n_feat -> h1
    hipMemsetAsync(agg, 0, (size_t)N * DIM * sizeof(float), stream);
    scatter_kernel<<<edgeBlocks32, edgeThreads, 0, stream>>>(n_feat, ew, src, dst, agg, E);
    gemm_concat_kernel<128><<<tileBlocks, 256, 0, stream>>>(agg, n_feat, inv, W1, b1, h1);

    // Layer 2: h1 -> h2
    hipMemsetAsync(agg, 0, (size_t)N * DIM * sizeof(float), stream);
    scatter_kernel<<<edgeBlocks32, edgeThreads, 0, stream>>>(h1, ew, src, dst, agg, E);
    gemm_concat_kernel<128><<<tileBlocks, 256, 0, stream>>>(agg, h1, inv, W2, b2, h2);

    // Layer 3: h2 -> out (64 cols)
    hipMemsetAsync(agg, 0, (size_t)N * DIM * sizeof(float), stream);
    scatter_kernel<<<edgeBlocks32, edgeThreads, 0, stream>>>(h2, ew, src, dst, agg, E);
    gemm_concat_kernel<64><<<tileBlocks, 128, 0, stream>>>(agg, h2, inv, W3, b3, (float*)d_out);
}